// SelfAttLayer_Dec_45165876085351
// MI455X (gfx1250) — compile-verified
//
#include <hip/hip_runtime.h>
#include <cstdint>

#define A_N 128
#define T_N 91
#define F_N 6
#define D_N 256
#define B_N (F_N * T_N)            // 546 time-batches
#define R_N ((size_t)B_N * A_N)    // 69888 attention rows
#define LDP 264                    // padded LDS row stride (floats) for 256-wide tiles
#define QKSTRIDE 72                // padded LDS stride for 64-wide Q/K/V tiles
#define PSTRIDE 136                // padded LDS stride for 128-wide P tile
#define QKOFF (128 * QKSTRIDE)     // K tile offset in attention LDS
#define VOFF (2 * QKOFF)           // V tile offset (above P reuse region)

typedef __attribute__((ext_vector_type(2))) float v2f;
typedef __attribute__((ext_vector_type(8))) float v8f;
typedef __attribute__((ext_vector_type(4))) unsigned int v4u;
typedef __attribute__((ext_vector_type(4))) int v4i;
typedef __attribute__((ext_vector_type(8))) int v8i;

__device__ __forceinline__ v8f wmma_f32(v2f a, v2f b, v8f c) {
  // D = A(16x4 f32) * B(4x16 f32) + C(16x16 f32), wave32
  return __builtin_amdgcn_wmma_f32_16x16x4_f32(false, a, false, b, (short)0, c,
                                               false, false);
}

// ---------------------------------------------------------------------------
// Tensor Data Mover: stage a 2D tile (rows x dim0 fp32, row stride stride0)
// from global memory into LDS at lds_byte_off, with optional hardware row
// padding (pad_interval_enc: 5 -> every 64 DW, 7 -> every 256 DW;
// pad_amount_enc: 7 -> insert 8 DW), per D# spec (CDNA5 ISA).
// ---------------------------------------------------------------------------
__device__ __forceinline__ void tdm_load_2d(uint32_t lds_byte_off,
                                            const float* gptr, uint32_t dim0,
                                            uint32_t rows, uint32_t stride0,
                                            uint32_t pad_interval_enc,
                                            uint32_t pad_amount_enc) {
  const uint64_t ga = (uint64_t)(uintptr_t)gptr;
  v4u g0;
  g0[0] = 1u;                                   // count=1, no gather
  g0[1] = lds_byte_off;                         // lds_addr
  g0[2] = (uint32_t)(ga & 0xFFFFFFFFu);         // global_addr[31:0]
  g0[3] = (uint32_t)((ga >> 32) & 0x1FFFFFFu)   // global_addr[56:32]
          | 0x80000000u;                        // type=2 ("image")
  v8i g1;
  uint32_t ctl = (2u << 16);                    // data_size = 4 bytes
  ctl |= (1u << 20) | (pad_interval_enc << 22) | (pad_amount_enc << 25);
  g1[0] = (int)ctl;                             // wg_mask=0, pad enabled
  g1[1] = (int)((dim0 & 0xFFFFu) << 16);        // tensor_dim0[15:0]
  g1[2] = (int)((dim0 >> 16) | ((rows & 0xFFFFu) << 16));  // dim0 hi, dim1 lo
  g1[3] = (int)((rows >> 16) | ((dim0 & 0xFFFFu) << 16));  // dim1 hi, tile_dim0
  g1[4] = (int)(rows & 0xFFFFu);                // tile_dim1, tile_dim2=0
  g1[5] = (int)stride0;                         // tensor_dim0_stride[31:0]
  g1[6] = 0;                                    // stride0 hi, stride1 lo
  g1[7] = 0;
  v4i g2 = {};
  v4i g3 = {};
#if defined(__clang_major__) && (__clang_major__ >= 23)
  v8i g4 = {};
  __builtin_amdgcn_tensor_load_to_lds(g0, g1, g2, g3, g4, 0);
#else
  __builtin_amdgcn_tensor_load_to_lds(g0, g1, g2, g3, 0);
#endif
}

// ---------------------------------------------------------------------------
// Kernel 1: LayerNorm1 + packed QKV projection.
// Block = 256 threads (8 waves), handles 64 rows of one time-batch b.
// ---------------------------------------------------------------------------
__global__ __launch_bounds__(256) void k1_ln_qkv(
    const float* __restrict__ x, const float* __restrict__ g1,
    const float* __restrict__ b1, const float* __restrict__ w_in,
    const float* __restrict__ b_in, float* __restrict__ qb,
    float* __restrict__ kb, float* __restrict__ vb) {
  __shared__ float Y[64 * LDP];
  const int blk = blockIdx.x;
  const int b = blk >> 1;
  const int a0 = (blk & 1) * 64;
  const int f = b / T_N, t = b % T_N;
  const int tid = threadIdx.x;
  const int w = tid >> 5, L = tid & 31;

  // ---- LayerNorm phase: wave w normalizes local rows 8w..8w+7 ----
  for (int lr = w * 8; lr < w * 8 + 8; ++lr) {
    const int a = a0 + lr;
    const float* xr = x + (((size_t)(f * A_N + a)) * T_N + t) * D_N;
    float v[8];
    float s = 0.f;
#pragma unroll
    for (int i = 0; i < 8; ++i) { v[i] = xr[i * 32 + L]; s += v[i]; }
    for (int m = 16; m >= 1; m >>= 1) s += __shfl_xor(s, m, 32);
    const float mean = s * (1.0f / 256.0f);
    float q = 0.f;
#pragma unroll
    for (int i = 0; i < 8; ++i) { float d = v[i] - mean; q += d * d; }
    for (int m = 16; m >= 1; m >>= 1) q += __shfl_xor(q, m, 32);
    const float rstd = rsqrtf(q * (1.0f / 256.0f) + 1e-5f);
#pragma unroll
    for (int i = 0; i < 8; ++i) {
      const int c = i * 32 + L;
      Y[lr * LDP + c] = (v[i] - mean) * rstd * g1[c] + b1[c];
    }
  }
  __syncthreads();

  // ---- GEMM phase: [64 x 256] x [256 x 768] ----
  const int mrow = 16 * (w >> 1);       // 4 row stripes
  const int chalf = 384 * (w & 1);      // 2 column halves of 768
  const int mlane = L & 15;
  const int kh2 = (L >> 4) * 2;
  const size_t rowg0 = (size_t)b * A_N + a0 + mrow;

  for (int chunk = 0; chunk < 3; ++chunk) {
    const int cbase = chalf + chunk * 128;
    v8f acc[8] = {};
    for (int k = 0; k < 256; k += 4) {
      const int kk = k + kh2;
      const v2f af = *(const v2f*)&Y[(mrow + mlane) * LDP + kk];
#pragma unroll
      for (int nt = 0; nt < 8; ++nt) {
        const int col = cbase + nt * 16 + mlane;
        const v2f bf = *(const v2f*)&w_in[(size_t)col * D_N + kk];
        acc[nt] = wmma_f32(af, bf, acc[nt]);
      }
    }
#pragma unroll
    for (int nt = 0; nt < 8; ++nt) {
      const int tb = cbase + nt * 16;   // wave-uniform tile column base
      float* dst;
      int coff;
      if (tb < 256)      { dst = qb; coff = tb; }
      else if (tb < 512) { dst = kb; coff = tb - 256; }
      else               { dst = vb; coff = tb - 512; }
      const int c = coff + mlane;
      const float bias = b_in[tb + mlane];
#pragma unroll
      for (int r = 0; r < 8; ++r) {
        const int mg = r + 8 * (L >> 4);
        dst[(rowg0 + mg) * D_N + c] = acc[nt][r] + bias;
      }
    }
  }
}

// ---------------------------------------------------------------------------
// Kernel 2: masked multi-head attention for one (b, h) pair.
// Q/K/V head slices staged by the Tensor Data Mover (hardware-padded rows),
// scores = QK^T/8, mask, softmax (incl. zero-attn key), out = P*V.
// Output overwrites the Q workspace slice in place.
// ---------------------------------------------------------------------------
__global__ __launch_bounds__(256) void k2_attn(
    const float* __restrict__ kbuf, const float* __restrict__ vbuf,
    float* __restrict__ qbuf, const unsigned char* __restrict__ batch_mask,
    const unsigned char* __restrict__ padding_mask) {
  __shared__ float lds[VOFF + 128 * QKSTRIDE];  // Q|K (-> P) | V
  const int bh = blockIdx.x;
  const int b = bh >> 2, h = bh & 3;
  const int t = b % T_N;
  const int tid = threadIdx.x, w = tid >> 5, L = tid & 31;
  const size_t rowbase = (size_t)b * A_N;
  const int coff = h * 64;
  const int mlane = L & 15, kh2 = (L >> 4) * 2;

  // ---- TDM-stage Q, K, V (128 x 64 each, stride 256 -> padded LDS 72) ----
  const uint32_t lds_base = (uint32_t)(uintptr_t)&lds[0];
  if (w == 0) {
    tdm_load_2d(lds_base, &qbuf[rowbase * D_N + coff], 64, 128, D_N, 5, 7);
    __builtin_amdgcn_s_wait_tensorcnt(0);
  } else if (w == 1) {
    tdm_load_2d(lds_base + QKOFF * 4, &kbuf[rowbase * D_N + coff], 64, 128,
                D_N, 5, 7);
    __builtin_amdgcn_s_wait_tensorcnt(0);
  } else if (w == 2) {
    tdm_load_2d(lds_base + VOFF * 4, &vbuf[rowbase * D_N + coff], 64, 128,
                D_N, 5, 7);
    __builtin_amdgcn_s_wait_tensorcnt(0);
  }
  __syncthreads();

  // ---- scores: wave w computes 16-row stripe x 128 cols ----
  v8f s[8] = {};
  for (int k = 0; k < 64; k += 4) {
    const int kk = k + kh2;
    const v2f af = *(const v2f*)&lds[(16 * w + mlane) * QKSTRIDE + kk];
#pragma unroll
    for (int nt = 0; nt < 8; ++nt) {
      const v2f bf = *(const v2f*)&lds[QKOFF + (nt * 16 + mlane) * QKSTRIDE + kk];
      s[nt] = wmma_f32(af, bf, s[nt]);
    }
  }

  // ---- mask + softmax in registers ----
  bool colpad[8];
#pragma unroll
  for (int nt = 0; nt < 8; ++nt)
    colpad[nt] = padding_mask[(nt * 16 + mlane) * T_N + t] != 0;

  const float NEGV = -1e30f;
#pragma unroll
  for (int r = 0; r < 8; ++r) {
    const int arow = 16 * w + r + 8 * (L >> 4);
    float mx = 0.0f;  // zero-attn column scores 0 and is never masked
#pragma unroll
    for (int nt = 0; nt < 8; ++nt) {
      const int j = nt * 16 + mlane;
      const bool msk = colpad[nt] || (batch_mask[arow * A_N + j] != 0);
      const float v = msk ? NEGV : s[nt][r] * 0.125f;  // 1/sqrt(64)
      s[nt][r] = v;
      mx = fmaxf(mx, v);
    }
    for (int m = 1; m < 16; m <<= 1) mx = fmaxf(mx, __shfl_xor(mx, m, 32));
    float sum = 0.f;
#pragma unroll
    for (int nt = 0; nt < 8; ++nt) {
      const float e = __expf(s[nt][r] - mx);
      s[nt][r] = e;
      sum += e;
    }
    sum += __expf(0.0f - mx);  // appended all-zero key
    for (int m = 1; m < 16; m <<= 1) sum += __shfl_xor(sum, m, 32);
    const float inv = 1.0f / sum;
#pragma unroll
    for (int nt = 0; nt < 8; ++nt) s[nt][r] *= inv;
  }

  // ---- spill P to LDS (reuse Q/K space) ----
  __syncthreads();
#pragma unroll
  for (int nt = 0; nt < 8; ++nt)
#pragma unroll
    for (int r = 0; r < 8; ++r)
      lds[(16 * w + r + 8 * (L >> 4)) * PSTRIDE + nt * 16 + mlane] = s[nt][r];
  __syncthreads();

  // ---- O = P (128x128) * V (128x64), V from LDS ----
  v8f o[4] = {};
  for (int k = 0; k < 128; k += 4) {
    const int kk = k + kh2;
    const v2f af = *(const v2f*)&lds[(16 * w + mlane) * PSTRIDE + kk];
#pragma unroll
    for (int nt = 0; nt < 4; ++nt) {
      const int n = nt * 16 + mlane;
      v2f bf;
      bf.x = lds[VOFF + (kk + 0) * QKSTRIDE + n];
      bf.y = lds[VOFF + (kk + 1) * QKSTRIDE + n];
      o[nt] = wmma_f32(af, bf, o[nt]);
    }
  }
#pragma unroll
  for (int nt = 0; nt < 4; ++nt) {
    const int n = nt * 16 + mlane;
#pragma unroll
    for (int r = 0; r < 8; ++r) {
      const int arow = 16 * w + r + 8 * (L >> 4);
      qbuf[(rowbase + arow) * D_N + coff + n] = o[nt][r];
    }
  }
}

// ---------------------------------------------------------------------------
// Kernel 3: out-projection + bias + residual, FFN (ReLU), LayerNorm2, store.
// Block handles 64 rows of one time-batch b (same mapping as kernel 1).
// ---------------------------------------------------------------------------
__global__ __launch_bounds__(256) void k3_proj_ffn_ln(
    const float* __restrict__ attb, const float* __restrict__ x,
    const float* __restrict__ w_out, const float* __restrict__ b_out,
    const float* __restrict__ w_f1, const float* __restrict__ b_f1,
    const float* __restrict__ g2, const float* __restrict__ b2,
    float* __restrict__ out) {
  __shared__ float Sm[64 * LDP];
  const int blk = blockIdx.x;
  const int b = blk >> 1, a0 = (blk & 1) * 64;
  const int f = b / T_N, t = b % T_N;
  const int tid = threadIdx.x, w = tid >> 5, L = tid & 31;
  const size_t rowg = (size_t)b * A_N + a0;

  // ---- TDM-stage attention tile (64 x 256, stride 256 -> padded 264) ----
  if (w == 0) {
    tdm_load_2d((uint32_t)(uintptr_t)&Sm[0], &attb[rowg * D_N], D_N, 64, D_N,
                7, 7);
    __builtin_amdgcn_s_wait_tensorcnt(0);
  }
  __syncthreads();

  const int mrow = 16 * (w >> 1);
  const int cb = 128 * (w & 1);
  const int mlane = L & 15, kh2 = (L >> 4) * 2;

  // ---- S = ATT * w_out^T + b_out + x ----
  v8f acc[8] = {};
  for (int k = 0; k < 256; k += 4) {
    const int kk = k + kh2;
    const v2f af = *(const v2f*)&Sm[(mrow + mlane) * LDP + kk];
#pragma unroll
    for (int nt = 0; nt < 8; ++nt) {
      const v2f bf = *(const v2f*)&w_out[(size_t)(cb + nt * 16 + mlane) * D_N + kk];
      acc[nt] = wmma_f32(af, bf, acc[nt]);
    }
  }
#pragma unroll
  for (int nt = 0; nt < 8; ++nt) {
    const int c = cb + nt * 16 + mlane;
    const float bias = b_out[c];
#pragma unroll
    for (int r = 0; r < 8; ++r) {
      const int a = a0 + mrow + r + 8 * (L >> 4);
      acc[nt][r] += bias + x[(((size_t)(f * A_N + a)) * T_N + t) * D_N + c];
    }
  }
  __syncthreads();  // all waves done reading ATT tile
#pragma unroll
  for (int nt = 0; nt < 8; ++nt)
#pragma unroll
    for (int r = 0; r < 8; ++r)
      Sm[(mrow + r + 8 * (L >> 4)) * LDP + cb + nt * 16 + mlane] = acc[nt][r];
  __syncthreads();

  // ---- F1 = relu(S * w_f1^T + b_f1) ----
  v8f acc2[8] = {};
  for (int k = 0; k < 256; k += 4) {
    const int kk = k + kh2;
    const v2f af = *(const v2f*)&Sm[(mrow + mlane) * LDP + kk];
#pragma unroll
    for (int nt = 0; nt < 8; ++nt) {
      const v2f bf = *(const v2f*)&w_f1[(size_t)(cb + nt * 16 + mlane) * D_N + kk];
      acc2[nt] = wmma_f32(af, bf, acc2[nt]);
    }
  }
  __syncthreads();  // all waves done reading S tile
#pragma unroll
  for (int nt = 0; nt < 8; ++nt) {
    const int c = cb + nt * 16 + mlane;
    const float bias = b_f1[c];
#pragma unroll
    for (int r = 0; r < 8; ++r)
      Sm[(mrow + r + 8 * (L >> 4)) * LDP + c] = fmaxf(acc2[nt][r] + bias, 0.0f);
  }
  __syncthreads();

  // ---- LayerNorm2 + store to [F, A, T, D] ----
  for (int lr = w * 8; lr < w * 8 + 8; ++lr) {
    float v[8];
    float s = 0.f;
#pragma unroll
    for (int i = 0; i < 8; ++i) { v[i] = Sm[lr * LDP + i * 32 + L]; s += v[i]; }
    for (int m = 16; m >= 1; m >>= 1) s += __shfl_xor(s, m, 32);
    const float mean = s * (1.0f / 256.0f);
    float q = 0.f;
#pragma unroll
    for (int i = 0; i < 8; ++i) { float d = v[i] - mean; q += d * d; }
    for (int m = 16; m >= 1; m >>= 1) q += __shfl_xor(q, m, 32);
    const float rstd = rsqrtf(q * (1.0f / 256.0f) + 1e-5f);
    const int a = a0 + lr;
    float* orow = out + (((size_t)(f * A_N + a)) * T_N + t) * D_N;
#pragma unroll
    for (int i = 0; i < 8; ++i) {
      const int c = i * 32 + L;
      orow[c] = (v[i] - mean) * rstd * g2[c] + b2[c];
    }
  }
}

// ---------------------------------------------------------------------------
extern "C" void kernel_launch(void* const* d_in, const int* in_sizes, int n_in,
                              void* d_out, int out_size, void* d_ws,
                              size_t ws_size, hipStream_t stream) {
  (void)in_sizes; (void)n_in; (void)out_size; (void)ws_size;
  const float* x     = (const float*)d_in[0];
  const float* ln1_g = (const float*)d_in[1];
  const float* ln1_b = (const float*)d_in[2];
  const float* w_in  = (const float*)d_in[3];
  const float* b_in  = (const float*)d_in[4];
  const float* w_out = (const float*)d_in[5];
  const float* b_out = (const float*)d_in[6];
  const float* w_f1  = (const float*)d_in[7];
  const float* b_f1  = (const float*)d_in[8];
  const float* ln2_g = (const float*)d_in[9];
  const float* ln2_b = (const float*)d_in[10];
  const unsigned char* batch_mask   = (const unsigned char*)d_in[11];
  const unsigned char* padding_mask = (const unsigned char*)d_in[12];
  float* out = (float*)d_out;

  float* ws = (float*)d_ws;
  float* qb = ws;                     // [R_N, 256]; later reused for att out
  float* kb = ws + R_N * D_N;         // [R_N, 256]
  float* vb = ws + 2 * R_N * D_N;     // [R_N, 256]

  const int blocks_rows = (int)(R_N / 64);  // 1092
  k1_ln_qkv<<<blocks_rows, 256, 0, stream>>>(x, ln1_g, ln1_b, w_in, b_in, qb,
                                             kb, vb);
  k2_attn<<<B_N * 4, 256, 0, stream>>>(kb, vb, qb, batch_mask, padding_mask);
  k3_proj_ffn_ln<<<blocks_rows, 256, 0, stream>>>(qb, x, w_out, b_out, w_f1,
                                                  b_f1, ln2_g, ln2_b, out);
}